// HiPPOLTI_42537356099800
// MI455X (gfx1250) — compile-verified
//
#include <hip/hip_runtime.h>

// ---------------------------------------------------------------------------
// HiPPO-LTI final-state via linear-recurrence reformulation (MI455X / gfx1250)
//
//   c_T[b] = Ad^4096 c0[b] + sum_k f[b,k] * (Ad^{4095-k} Bd)
//
// Stage 1: forward substitution (part1 = I - 0.5A is lower triangular)
// Stage 2: seed V rows 4080..4095 serially, then log-depth block doubling
//          using fp32 WMMA GEMMs (V block_new = V block_old * G^T, G = Ad^{2^j})
// Stage 3: Out = f * V (split-K=4 across grid.z) + c0 * (Ad^4096)^T + reduce
//
// GEMM: one wave32 computes a 16x64 C tile: 1 shared A-fragment, 4 B-fragments,
// 4 independent V_WMMA_F32_16X16X4_F32 accumulator chains; explicit register
// double-buffering so loads for chunk k+4 are in flight during wmma of chunk k.
// Workspace: 3*256*256 + 256 + 4096*256 + 3*512*256 floats ~= 6.3 MB.
// ---------------------------------------------------------------------------

typedef __attribute__((ext_vector_type(2))) float v2f;
typedef __attribute__((ext_vector_type(8))) float v8f;

#define N_ORDER 256
#define BATCH   512
#define SEQ_LEN 4096

// ---- Stage 1: column-parallel forward substitution --------------------------
__global__ void hippo_solve_kernel(const float* __restrict__ A,
                                   const float* __restrict__ Bm,
                                   float* __restrict__ Ad,
                                   float* __restrict__ G,
                                   float* __restrict__ Bd) {
    int c = threadIdx.x;
    if (c > N_ORDER) return;           // threads 257..287 idle
    float x[N_ORDER];                  // private column history (scratch)
    for (int i = 0; i < N_ORDER; ++i) {
        float s;
        if (c < N_ORDER) s = (i == c ? 1.0f : 0.0f) + 0.5f * A[i * N_ORDER + c];
        else             s = Bm[i];
        for (int j = 0; j < i; ++j)
            s += 0.5f * A[i * N_ORDER + j] * x[j];   // s -= part1[i][j]*x[j]
        float xi = s / (1.0f - 0.5f * A[i * N_ORDER + i]);
        x[i] = xi;
        if (c < N_ORDER) { Ad[i * N_ORDER + c] = xi; G[i * N_ORDER + c] = xi; }
        else               Bd[i] = xi;
    }
}

// ---- Stage 2 seed: V[4095-p] = Ad^p Bd for p = 0..15 (serial matvec) -------
__global__ void hippo_seed_kernel(const float* __restrict__ Ad,
                                  const float* __restrict__ Bd,
                                  float* __restrict__ V) {
    __shared__ float prev[N_ORDER];
    __shared__ float nxt[N_ORDER];
    int m = threadIdx.x;
    prev[m] = Bd[m];
    __syncthreads();
    for (int p = 0; p < 16; ++p) {
        V[(size_t)(4095 - p) * N_ORDER + m] = prev[m];
        float s = 0.0f;
        for (int n = 0; n < N_ORDER; ++n)
            s += Ad[m * N_ORDER + n] * prev[n];
        nxt[m] = s;
        __syncthreads();
        prev[m] = nxt[m];
        __syncthreads();
    }
}

// ---- fragment loader (straight-line per TRANSB specialization) --------------
template <int TRANSB>
__device__ __forceinline__ void load_frags(const float* __restrict__ arow,
                                           const float* __restrict__ brow0,
                                           const float* __restrict__ bcol,
                                           int ldb, int k0,
                                           v2f& a, v2f* b) {
    a.x = arow[k0];
    a.y = arow[k0 + 1];
#pragma unroll
    for (int t = 0; t < 4; ++t) {
        if (TRANSB) {
            const float* bp = brow0 + (size_t)(16 * t) * ldb + k0;
            b[t].x = bp[0];
            b[t].y = bp[1];
        } else {
            const float* bp = bcol + (size_t)k0 * ldb + 16 * t;
            b[t].x = bp[0];
            b[t].y = bp[ldb];
        }
    }
}

// ---- fp32 WMMA GEMM: C(MxN) = A(MxK) * op(B) [+ C], optional split-K --------
// One wave32 per 16x64 C tile. grid = (N/64, M/16, splits), block = 32.
// grid.z slice z covers K range [z*kPerSplit, (z+1)*kPerSplit); z==0 writes
// Cmat, z>0 writes Cpart + (z-1)*partStride (reduced later).
// ISA layouts (CDNA5 ISA 7.12.2), wave32:
//   A 16x4:  lane L -> m = L%16, K pair base = 2*(L/16)
//   B 4x16:  lane L -> n = L%16, K pair base = 2*(L/16)
//   C 16x16: VGPR v, lane L -> row = v + 8*(L/16), col = L%16
template <int TRANSB, int ACCUM>
__global__ void gemm_wmma_f32(const float* __restrict__ Amat,
                              const float* __restrict__ Bmat,
                              float* __restrict__ Cmat,
                              int kPerSplit, int lda, int ldb, int ldc,
                              float* __restrict__ Cpart, size_t partStride) {
    const int lane = threadIdx.x;      // 0..31 (wave32)
    const int m    = lane & 15;
    const int half = lane >> 4;        // 0 or 1
    const int tm   = blockIdx.y << 4;
    const int tn0  = blockIdx.x << 6;  // 64-wide N tile
    const int z    = blockIdx.z;
    const int kb   = z * kPerSplit;

    float* Cout = (z == 0) ? Cmat : (Cpart + (size_t)(z - 1) * partStride);

    v8f acc[4] = {};
    if (ACCUM) {
#pragma unroll
        for (int t = 0; t < 4; ++t)
#pragma unroll
            for (int v = 0; v < 8; ++v)
                acc[t][v] = Cout[(size_t)(tm + v + 8 * half) * ldc + tn0 + 16 * t + m];
    }

    const float* arow  = Amat + (size_t)(tm + m) * lda + kb + 2 * half;
    const float* brow0 = Bmat + (size_t)(tn0 + m) * ldb + kb + 2 * half;       // TRANSB
    const float* bcol  = Bmat + (size_t)(kb + 2 * half) * ldb + tn0 + m;       // !TRANSB

    // software pipeline: loads for chunk k+4 issue before wmma of chunk k
    v2f a_cur, b_cur[4];
    load_frags<TRANSB>(arow, brow0, bcol, ldb, 0, a_cur, b_cur);

#pragma unroll 4
    for (int k0 = 0; k0 + 4 < kPerSplit; k0 += 4) {
        v2f a_nxt, b_nxt[4];
        load_frags<TRANSB>(arow, brow0, bcol, ldb, k0 + 4, a_nxt, b_nxt);
        __builtin_prefetch(arow + k0 + 512, 0, 3);   // near-cache prefetch

#pragma unroll
        for (int t = 0; t < 4; ++t)
            acc[t] = __builtin_amdgcn_wmma_f32_16x16x4_f32(false, a_cur, false, b_cur[t],
                                                           (short)0, acc[t],
                                                           false, false);
        a_cur = a_nxt;
#pragma unroll
        for (int t = 0; t < 4; ++t) b_cur[t] = b_nxt[t];
    }
#pragma unroll
    for (int t = 0; t < 4; ++t)
        acc[t] = __builtin_amdgcn_wmma_f32_16x16x4_f32(false, a_cur, false, b_cur[t],
                                                       (short)0, acc[t],
                                                       false, false);

#pragma unroll
    for (int t = 0; t < 4; ++t)
#pragma unroll
        for (int v = 0; v < 8; ++v)
            Cout[(size_t)(tm + v + 8 * half) * ldc + tn0 + 16 * t + m] = acc[t][v];
}

// ---- split-K reduction: out[i] += P0[i] + P1[i] + P2[i] ---------------------
__global__ void reduce_add3_kernel(float* __restrict__ out,
                                   const float* __restrict__ p,
                                   size_t n, size_t stride) {
    size_t i = (size_t)blockIdx.x * blockDim.x + threadIdx.x;
    if (i < n)
        out[i] += p[i] + p[i + stride] + p[i + 2 * stride];
}

// ---------------------------------------------------------------------------
extern "C" void kernel_launch(void* const* d_in, const int* in_sizes, int n_in,
                              void* d_out, int out_size, void* d_ws, size_t ws_size,
                              hipStream_t stream) {
    const float* f  = (const float*)d_in[0];   // (512, 4096)
    const float* A  = (const float*)d_in[1];   // (256, 256)
    const float* Bm = (const float*)d_in[2];   // (256, 1)
    const float* c0 = (const float*)d_in[3];   // (512, 1, 256)
    float* out = (float*)d_out;                // (512, 1, 256)

    float* ws = (float*)d_ws;                  // needs ~6.3 MB
    float* Ad = ws;                            // 256*256
    float* G0 = Ad + 256 * 256;                // 256*256 (power ping)
    float* G1 = G0 + 256 * 256;                // 256*256 (power pong)
    float* Bd = G1 + 256 * 256;                // 256
    float* V  = Bd + 256;                      // 4096*256: V[k] = Ad^{4095-k} Bd
    float* P  = V + (size_t)4096 * 256;        // 3 * 512*256 split-K partials
    const size_t PSTRIDE = (size_t)BATCH * N_ORDER;   // 131072

    // Stage 1: Ad, Bd (and G0 = Ad copy) by forward substitution.
    hipLaunchKernelGGL(hippo_solve_kernel, dim3(1), dim3(288), 0, stream,
                       A, Bm, Ad, G0, Bd);

    // Stage 2 seed: V rows 4080..4095 (powers 0..15).
    hipLaunchKernelGGL(hippo_seed_kernel, dim3(1), dim3(N_ORDER), 0, stream,
                       Ad, Bd, V);

    float* gcur = G0;
    float* gnxt = G1;
    // Square Ad four times: gcur -> Ad^16.
    for (int i = 0; i < 4; ++i) {
        hipLaunchKernelGGL((gemm_wmma_f32<0, 0>), dim3(4, 16, 1), dim3(32), 0, stream,
                           gcur, gcur, gnxt, 256, 256, 256, 256, (float*)nullptr, (size_t)0);
        float* t = gcur; gcur = gnxt; gnxt = t;
    }
    // Block doubling: rows [4096-2s, 4096-s) = rows [4096-s, 4096) * G^T.
    for (int s = 16; s <= 2048; s <<= 1) {
        hipLaunchKernelGGL((gemm_wmma_f32<1, 0>), dim3(4, s / 16, 1), dim3(32), 0, stream,
                           V + (size_t)(4096 - s) * 256, gcur,
                           V + (size_t)(4096 - 2 * s) * 256,
                           256, 256, 256, 256, (float*)nullptr, (size_t)0);
        hipLaunchKernelGGL((gemm_wmma_f32<0, 0>), dim3(4, 16, 1), dim3(32), 0, stream,
                           gcur, gcur, gnxt, 256, 256, 256, 256, (float*)nullptr, (size_t)0);
        float* t = gcur; gcur = gnxt; gnxt = t;
    }
    // gcur == Ad^4096.

    // Stage 3a: Out/partials = f * V, split-K=4 (M=512, N=256, K=4*1024)
    hipLaunchKernelGGL((gemm_wmma_f32<0, 0>), dim3(4, 32, 4), dim3(32), 0, stream,
                       f, V, out, 1024, 4096, 256, 256, P, PSTRIDE);
    // Stage 3b: Out += c0 * (Ad^4096)^T  (M=512, N=256, K=256)
    hipLaunchKernelGGL((gemm_wmma_f32<1, 1>), dim3(4, 32, 1), dim3(32), 0, stream,
                       c0, gcur, out, 256, 256, 256, 256, (float*)nullptr, (size_t)0);
    // Stage 3c: Out += P0 + P1 + P2
    hipLaunchKernelGGL(reduce_add3_kernel, dim3((unsigned)(PSTRIDE / 256)), dim3(256),
                       0, stream, out, P, PSTRIDE, PSTRIDE);

    (void)in_sizes; (void)n_in; (void)out_size; (void)ws_size;
}